// modified_with_attention_mask_77154792506036
// MI455X (gfx1250) — compile-verified
//
#include <hip/hip_runtime.h>
#include <hip/hip_bf16.h>
#include <math.h>

// Problem constants (match reference)
#define BSZ   256
#define NTOK  196
#define EDIM  768
#define MLP   2
#define ALPHA 0.2f
#define BN    (BSZ * NTOK)        // 50176 rows, divisible by 16

typedef __attribute__((ext_vector_type(2))) float v2f;
typedef __attribute__((ext_vector_type(8))) float v8f;

// ---------------------------------------------------------------------------
// Kernel 1: Wh[b,n] = sent_emb[b,n,:] . W_w[:] + W_b   via V_WMMA_F32_16X16X4_F32
// One wave per 16-row strip of the [BN, E] matrix. B-matrix columns all hold
// W_w[k] (broadcast), so each accumulator column is the row dot-product.
// A-frag (16x4 f32): lane m=l&15, half=l>>4 -> K = {2*half, 2*half+1}  (v2f, 8B load)
// B-frag (4x16 f32): same K striping per half-wave, value = w[k] for every column.
// D (16x16 f32): lanes 0-15 hold M=v, lanes 16-31 hold M=8+v -> lanes 0 and 16 store.
// ---------------------------------------------------------------------------
__global__ __launch_bounds__(256)
void wh_wmma_kernel(const float* __restrict__ X,     // [BN, E]
                    const float* __restrict__ W_w,   // [E]
                    const float* __restrict__ W_b,   // [1]
                    float* __restrict__ wh)          // [BN]
{
    __shared__ float w_lds[EDIM];
    for (int i = threadIdx.x; i < EDIM; i += blockDim.x) w_lds[i] = W_w[i];
    __syncthreads();

    const int wave = threadIdx.x >> 5;
    const int lane = threadIdx.x & 31;
    const int tile = blockIdx.x * 8 + wave;          // 16-row strip index
    const int mbase = tile * 16;
    if (mbase >= BN) return;

    const int m    = lane & 15;
    const int half = lane >> 4;                      // 0: K pair {0,1}, 1: {2,3}

    const float* arow = X + (size_t)(mbase + m) * EDIM + half * 2;
    const float* brow = w_lds + half * 2;

    v8f c = {};
    #pragma unroll 8
    for (int k = 0; k < EDIM; k += 4) {
        v2f a = *(const v2f*)(arow + k);             // aligned 8B global load
        v2f b = *(const v2f*)(brow + k);             // aligned ds_load_b64
        c = __builtin_amdgcn_wmma_f32_16x16x4_f32(
                /*neg_a=*/false, a, /*neg_b=*/false, b,
                /*c_mod=*/(short)0, c, /*reuse_a=*/false, /*reuse_b=*/false);
    }

    if ((lane & 15) == 0) {                          // lanes 0 (rows 0-7) and 16 (rows 8-15)
        const float bb = W_b[0];
        const int r = mbase + half * 8;
        #pragma unroll
        for (int v = 0; v < 8; ++v) wh[r + v] = c[v] + bb;
    }
}

// ---------------------------------------------------------------------------
// Kernel 2: span mean pooling -> pair[b, 0:768]=mean(span1), pair[b,768:1536]=mean(span2)
// One block per batch element; threads stride over E. Reads only span rows (~5 MB total).
// ---------------------------------------------------------------------------
__global__ __launch_bounds__(256)
void span_mean_kernel(const float* __restrict__ X,       // [B, N, E]
                      const int*   __restrict__ batch_e, // [B, 4]
                      float* __restrict__ pair)          // [B, 2E]
{
    const int b  = blockIdx.x;
    const int s1 = batch_e[b * 4 + 0], t1 = batch_e[b * 4 + 1];
    const int s2 = batch_e[b * 4 + 2], t2 = batch_e[b * 4 + 3];
    const float inv1 = 1.0f / (float)(t1 - s1);
    const float inv2 = 1.0f / (float)(t2 - s2);
    const float* Xb = X + (size_t)b * NTOK * EDIM;

    for (int e = threadIdx.x; e < EDIM; e += blockDim.x) {
        float a1 = 0.f, a2 = 0.f;
        for (int n = s1; n < t1; ++n) a1 += Xb[(size_t)n * EDIM + e];
        for (int n = s2; n < t2; ++n) a2 += Xb[(size_t)n * EDIM + e];
        pair[(size_t)b * (2 * EDIM) + e]        = a1 * inv1;
        pair[(size_t)b * (2 * EDIM) + EDIM + e] = a2 * inv2;
    }
}

// ---------------------------------------------------------------------------
// Kernel 3: per-batch GAT softmax (attn_w>0 mask is provably all-true), elu,
// sent/event heads, 0.85/0.15 mix. One block per batch element.
// ---------------------------------------------------------------------------
__global__ __launch_bounds__(256)
void finalize_kernel(const float* __restrict__ wh,       // [B*N]
                     const float* __restrict__ pair,     // [B, 2E]
                     const float* __restrict__ aT_w,     // [2]
                     const float* __restrict__ event_W,  // [MLP, 2E]
                     const float* __restrict__ event_b,  // [MLP]
                     const float* __restrict__ sent_W,   // [MLP, N]
                     const float* __restrict__ sent_b,   // [MLP]
                     float* __restrict__ out)            // [B, MLP]
{
    __shared__ float wh_s[NTOK];
    __shared__ float emb_s[NTOK];
    __shared__ float red[4][256];

    const int b = blockIdx.x;
    const int t = threadIdx.x;

    for (int i = t; i < NTOK; i += blockDim.x) wh_s[i] = wh[(size_t)b * NTOK + i];
    __syncthreads();

    const float a0 = aT_w[0], a1 = aT_w[1];

    if (t < NTOK) {
        const float whi = a0 * wh_s[t];
        // pass 1: row max of leaky_relu(a0*wh_i + a1*wh_j)
        float mx = -INFINITY;
        for (int j = 0; j < NTOK; ++j) {
            float x = whi + a1 * wh_s[j];
            x = (x > 0.f) ? x : ALPHA * x;
            mx = fmaxf(mx, x);
        }
        // pass 2: softmax-weighted sum of Wh
        float sum = 0.f, dot = 0.f;
        for (int j = 0; j < NTOK; ++j) {
            float x = whi + a1 * wh_s[j];
            x = (x > 0.f) ? x : ALPHA * x;
            float p = __expf(x - mx);
            sum += p;
            dot += p * wh_s[j];
        }
        float h = dot / sum;
        emb_s[t] = (h > 0.f) ? h : expm1f(h);        // elu
    }
    __syncthreads();

    // partial dots: event head (2E=1536) and sentence head (N=196)
    float p0 = 0.f, p1 = 0.f, s0 = 0.f, s1 = 0.f;
    const float* pb = pair + (size_t)b * (2 * EDIM);
    for (int c = t; c < 2 * EDIM; c += blockDim.x) {
        float pv = pb[c];
        p0 += pv * event_W[c];
        p1 += pv * event_W[2 * EDIM + c];
    }
    for (int i = t; i < NTOK; i += blockDim.x) {
        float ev = emb_s[i];
        s0 += ev * sent_W[i];
        s1 += ev * sent_W[NTOK + i];
    }
    red[0][t] = p0; red[1][t] = p1; red[2][t] = s0; red[3][t] = s1;
    __syncthreads();
    for (int off = 128; off > 0; off >>= 1) {
        if (t < off) {
            red[0][t] += red[0][t + off];
            red[1][t] += red[1][t + off];
            red[2][t] += red[2][t + off];
            red[3][t] += red[3][t + off];
        }
        __syncthreads();
    }
    if (t == 0) {
        float ev0 = red[0][0] + event_b[0];
        float ev1 = red[1][0] + event_b[1];
        float sp0 = red[2][0] + sent_b[0];
        float sp1 = red[3][0] + sent_b[1];
        out[b * MLP + 0] = 0.85f * ev0 + 0.15f * sp0;
        out[b * MLP + 1] = 0.85f * ev1 + 0.15f * sp1;
    }
}

// ---------------------------------------------------------------------------
// Input order: 0 sent_emb, 1 batch_e, 2 Wq1, 3 bq1, 4 Wk1, 5 bk1, 6 Wq2, 7 bq2,
// 8 Wk2, 9 bk2, 10 W_w, 11 W_b, 12 aT_w, 13 event_W, 14 event_b, 15 sent_W, 16 sent_b
// (Wq*/Wk*/bq*/bk* feed only attn_w>0, which is always true -> dead code.)
// ---------------------------------------------------------------------------
extern "C" void kernel_launch(void* const* d_in, const int* in_sizes, int n_in,
                              void* d_out, int out_size, void* d_ws, size_t ws_size,
                              hipStream_t stream) {
    const float* sent_emb = (const float*)d_in[0];
    const int*   batch_e  = (const int*)  d_in[1];
    const float* W_w      = (const float*)d_in[10];
    const float* W_b      = (const float*)d_in[11];
    const float* aT_w     = (const float*)d_in[12];
    const float* event_W  = (const float*)d_in[13];
    const float* event_b  = (const float*)d_in[14];
    const float* sent_W   = (const float*)d_in[15];
    const float* sent_b   = (const float*)d_in[16];
    float*       out      = (float*)d_out;

    float* wh_ws   = (float*)d_ws;                         // [BN]       200 KB
    float* pair_ws = wh_ws + BN;                           // [B, 2E]    1.5 MB

    // 3136 strips of 16 rows, 8 waves (strips) per 256-thread block
    wh_wmma_kernel<<<(BN / 16 + 7) / 8, 256, 0, stream>>>(sent_emb, W_w, W_b, wh_ws);
    span_mean_kernel<<<BSZ, 256, 0, stream>>>(sent_emb, batch_e, pair_ws);
    finalize_kernel<<<BSZ, 256, 0, stream>>>(wh_ws, pair_ws, aT_w, event_W, event_b,
                                             sent_W, sent_b, out);
}